// max_spherical_wassersten_distance_Residual_56642028699971
// MI455X (gfx1250) — compile-verified
//
#include <hip/hip_runtime.h>
#include <math.h>

// ---------------------------------------------------------------------------
// Sliced spherical Wasserstein (circular W1, level-median form), MI455X/gfx1250.
// One workgroup per projection; all intermediate state in LDS (CDNA5: 320KB/WGP).
// Projection GEMM ([4096x3]x[3x2]) done with V_WMMA_F32_16X16X4_F32, two tiles
// per wave iteration so the atan2 polynomial runs with all 32 lanes productive.
// ---------------------------------------------------------------------------

typedef __attribute__((ext_vector_type(2))) float v2f;
typedef __attribute__((ext_vector_type(8))) float v8f;

#define NPTS    4096          // points per cloud (n == m)
#define NMERGE  8192          // merged size
#define NBINS   8193          // integer cdf-diff numerators in [-4096, 4096]
#define NTHR    256           // 8 wave32
#define NWAVES  8
#define NTILES  (NPTS / 16)
#define CHUNK   33            // ceil(8193/256)

__device__ __forceinline__ float angle01(float xp, float yp) {
    // atan2 is scale invariant, so the reference's normalize step is a no-op here.
    const float PI = 3.14159265358979323846f;
    return (atan2f(-yp, -xp) + PI) * (1.0f / (2.0f * PI));
}

__global__ void zero_out_kernel(float* out, int n) {
    int i = blockIdx.x * blockDim.x + threadIdx.x;
    if (i < n) out[i] = 0.0f;
}

extern "C" __global__ __launch_bounds__(NTHR)
void ssw_circle_w1_kernel(const float* __restrict__ Xs,
                          const float* __restrict__ Xt,
                          const float* __restrict__ Us,
                          float* __restrict__ out,
                          int P, float invP)
{
    extern __shared__ char smem_raw[];
    float* s_u    = (float*)smem_raw;               // [4096] angles src (then sorted)
    float* s_v    = s_u + NPTS;                     // [4096] angles tgt (then sorted)
    float* s_vals = s_v + NPTS;                     // [8193] merged values (+1.0 pad)
    int*   s_cdf  = (int*)(s_vals + NMERGE + 1);    // [8192] cdf-diff numerator
    float* s_hist = (float*)(s_cdf + NMERGE);       // [8193] delta-weight histogram
    float* s_ct   = s_hist + NBINS;                 // [8][2][16][16] per-wave C tiles
    float* s_part = s_ct + NWAVES * 512;            // [256] chunk sums -> excl prefix
    float* s_red  = s_part + NTHR;                  // [256] final reduction
    int*   s_lev  = (int*)(s_red + NTHR);           // [1] level-median numerator

    const int tid  = threadIdx.x;
    const int lane = tid & 31;
    const int wave = tid >> 5;
    const int p    = blockIdx.x;
    if (p >= P) return;

    // ---- B operand: U[3][2] embedded in a 4x16 f32 matrix (cols 0,1; K=3 padded) ----
    const float* U = Us + (size_t)p * 6;            // Us[p][d][k], row-major
    v2f bmat; bmat.x = 0.f; bmat.y = 0.f;
    {
        int nn = lane & 15;
        if (nn < 2) {
            if (lane < 16) { bmat.x = U[0 * 2 + nn]; bmat.y = U[1 * 2 + nn]; } // K=0,1
            else           { bmat.x = U[2 * 2 + nn]; bmat.y = 0.f;           } // K=2,3
        }
    }

    // ---- Phase 1: project 32 points (2 tiles) per wave-iter with WMMA, then angles ----
    const bool hi = (lane >= 16);
    const int  ml = lane & 15;
    for (int src = 0; src < 2; ++src) {
        const float* X = (src == 0) ? Xs : Xt;
        float* sang    = (src == 0) ? s_u : s_v;
        for (int tt = wave; tt < NTILES / 2; tt += NWAVES) {  // uniform 16 pairs/wave
            const int t0 = tt * 2;
            // Branchless A operands (each lane loads its point's x,y,z; cndmask-select)
            const float* xr0 = X + (size_t)(t0 * 16 + ml) * 3;
            const float* xr1 = X + (size_t)(t0 * 16 + 16 + ml) * 3;
            float x0 = xr0[0], y0 = xr0[1], z0 = xr0[2];
            float x1 = xr1[0], y1 = xr1[1], z1 = xr1[2];
            v2f a0, a1;
            a0.x = hi ? z0 : x0;  a0.y = hi ? 0.f : y0;   // lanes<16: K=0,1 ; >=16: K=2,3(pad)
            a1.x = hi ? z1 : x1;  a1.y = hi ? 0.f : y1;
            v8f c0 = {0.f,0.f,0.f,0.f,0.f,0.f,0.f,0.f};
            v8f c1 = {0.f,0.f,0.f,0.f,0.f,0.f,0.f,0.f};
            c0 = __builtin_amdgcn_wmma_f32_16x16x4_f32(false, a0, false, bmat,
                                                       (short)0, c0, false, false);
            c1 = __builtin_amdgcn_wmma_f32_16x16x4_f32(false, a1, false, bmat,
                                                       (short)0, c1, false, false);
            // Scatter C tiles: VGPR r, lane l -> M = r + 8*(l/16), N = l%16
            float* ct = s_ct + wave * 512;
            const int half = (lane >> 4) * 8;
#pragma unroll
            for (int r = 0; r < 8; ++r) {
                ct[(half + r) * 16 + ml]       = c0[r];
                ct[256 + (half + r) * 16 + ml] = c1[r];
            }
            __builtin_amdgcn_wave_barrier();   // same-wave LDS ops are in-order
            // All 32 lanes productive: lanes<16 -> tile t0, lanes>=16 -> tile t0+1
            const int sel = lane >> 4;
            float xp = ct[sel * 256 + ml * 16 + 0];
            float yp = ct[sel * 256 + ml * 16 + 1];
            sang[(t0 + sel) * 16 + ml] = angle01(xp, yp);
            __builtin_amdgcn_wave_barrier();
        }
    }
    __syncthreads();

    // ---- Phase 2: bitonic sort both 4096-element angle arrays in LDS ----
    for (int k = 2; k <= NPTS; k <<= 1) {
        for (int j = k >> 1; j > 0; j >>= 1) {
            for (int idx = tid; idx < NPTS; idx += NTHR) {
                int ixj = idx ^ j;
                if (ixj > idx) {
                    bool up = ((idx & k) == 0);
                    float a = s_u[idx], b = s_u[ixj];
                    if ((a > b) == up) { s_u[idx] = b; s_u[ixj] = a; }
                    float e = s_v[idx], f = s_v[ixj];
                    if ((e > f) == up) { s_v[idx] = f; s_v[ixj] = e; }
                }
            }
            __syncthreads();
        }
    }

    // ---- Phase 3: merge by binary search; integer cdf-diff numerators ----
    // u element i -> merged pos m = i + #{v < u[i]} ; cdf*n = 2i+1-m
    // v element j -> merged pos m = j + #{u <= v[j]}; cdf*n = m-2j-1  (stable tie order)
    for (int i = tid; i < NPTS; i += NTHR) {
        float key = s_u[i];
        int lo = 0, hiB = NPTS;
        while (lo < hiB) { int mid = (lo + hiB) >> 1; if (s_v[mid] < key) lo = mid + 1; else hiB = mid; }
        int m = i + lo;
        s_vals[m] = key;
        s_cdf[m]  = 2 * i + 1 - m;

        key = s_v[i];
        lo = 0; hiB = NPTS;
        while (lo < hiB) { int mid = (lo + hiB) >> 1; if (s_u[mid] <= key) lo = mid + 1; else hiB = mid; }
        m = i + lo;
        s_vals[m] = key;
        s_cdf[m]  = m - 2 * i - 1;
    }
    for (int b = tid; b < NBINS; b += NTHR) s_hist[b] = 0.f;
    if (tid == 0) { s_vals[NMERGE] = 1.0f; *s_lev = 0; }
    __syncthreads();

    // ---- Phase 4: histogram delta weights by cdf-diff bin (ds_add_f32) ----
    for (int m = tid; m < NMERGE; m += NTHR) {
        float delta = s_vals[m + 1] - s_vals[m];
        atomicAdd(&s_hist[s_cdf[m] + NPTS], delta);
    }
    __syncthreads();

    // ---- Phase 5: weighted level-median = first bin where cum weight >= 0.5 ----
    {
        int base = tid * CHUNK;
        float s = 0.f;
        for (int b2 = 0; b2 < CHUNK; ++b2) {
            int bi = base + b2;
            if (bi < NBINS) s += s_hist[bi];
        }
        s_part[tid] = s;
    }
    __syncthreads();
    if (tid == 0) {
        float run = 0.f;
        for (int c2 = 0; c2 < NTHR; ++c2) { float t2 = s_part[c2]; s_part[c2] = run; run += t2; }
    }
    __syncthreads();
    {
        float pre = s_part[tid];
        if (pre < 0.5f) {                 // only the crossing chunk will find one
            float run = pre;
            int base = tid * CHUNK, found = -1;
            for (int b2 = 0; b2 < CHUNK && found < 0; ++b2) {
                int bi = base + b2;
                if (bi < NBINS) {
                    run += s_hist[bi];
                    if (run >= 0.5f) found = bi;
                }
            }
            if (found >= 0) *s_lev = found - NPTS;
        }
    }
    __syncthreads();

    // ---- Phase 6: W1 = sum delta * |cdf - levMed| ; block reduce ; atomicAdd mean ----
    const int lev = *s_lev;
    float acc = 0.f;
    for (int m = tid; m < NMERGE; m += NTHR) {
        float delta = s_vals[m + 1] - s_vals[m];
        acc += delta * fabsf((float)(s_cdf[m] - lev));
    }
    acc *= (1.0f / (float)NPTS);
    s_red[tid] = acc;
    __syncthreads();
    for (int off = NTHR >> 1; off > 0; off >>= 1) {
        if (tid < off) s_red[tid] += s_red[tid + off];
        __syncthreads();
    }
    if (tid == 0) atomicAdd(out, s_red[0] * invP);
}

extern "C" void kernel_launch(void* const* d_in, const int* in_sizes, int n_in,
                              void* d_out, int out_size, void* d_ws, size_t ws_size,
                              hipStream_t stream) {
    const float* Xs = (const float*)d_in[0];
    const float* Xt = (const float*)d_in[1];
    const float* Us = (const float*)d_in[2];
    float* out = (float*)d_out;
    const int P = in_sizes[2] / 6;   // Us: [P,3,2]

    zero_out_kernel<<<dim3((out_size + 255) / 256), dim3(256), 0, stream>>>(out, out_size);

    const size_t smem_elems = (size_t)NPTS + NPTS + (NMERGE + 1) + NMERGE + NBINS
                            + (size_t)NWAVES * 512 + NTHR + NTHR + 2;
    const size_t smem_bytes = smem_elems * 4;   // ~149 KB, fits CDNA5 320 KB/WGP
    ssw_circle_w1_kernel<<<dim3(P), dim3(NTHR), smem_bytes, stream>>>(
        Xs, Xt, Us, out, P, 1.0f / (float)P);
}